// GraphAttentionLayer_2413771620834
// MI455X (gfx1250) — compile-verified
//
#include <hip/hip_runtime.h>

typedef float v2f __attribute__((ext_vector_type(2)));
typedef float v8f __attribute__((ext_vector_type(8)));

#define B_ 4
#define N_ 4096
#define F_ 64

// Workspace layout (floats):
//   Wh    [B,N,F]   at 0
//   s1    [B,N]     at S1_OFF
//   s2    [B,N]     at S2_OFF
//   s2max [B]       at S2MAX_OFF
static constexpr long WH_ELEMS  = (long)B_ * N_ * F_;     // 1,048,576
static constexpr long S1_OFF    = WH_ELEMS;
static constexpr long S2_OFF    = S1_OFF + (long)B_ * N_;
static constexpr long S2MAX_OFF = S2_OFF + (long)B_ * N_;

#if __has_builtin(__builtin_amdgcn_exp2f)
#define EXP2F(x) __builtin_amdgcn_exp2f(x)
#else
#define EXP2F(x) exp2f(x)
#endif

// ---------------------------------------------------------------------------
// Kernel 1: Wh = h @ W  (per 4 rows / block), plus s1 = Wh.a[:F], s2 = Wh.a[F:]
// ---------------------------------------------------------------------------
__global__ void __launch_bounds__(256)
gat_wh_kernel(const float* __restrict__ h, const float* __restrict__ W,
              const float* __restrict__ a, float* __restrict__ ws) {
    __shared__ float Wl[F_ * F_];      // 16 KB
    __shared__ float hl[4][F_];
    __shared__ float al[2 * F_];
    __shared__ float red1[4][F_];
    __shared__ float red2[4][F_];

    const int t = threadIdx.x;
    for (int i = t; i < F_ * F_; i += 256) Wl[i] = W[i];
    if (t < 2 * F_) al[t] = a[t];

    const long base = (long)blockIdx.x * 4;   // flattened (b*N+n) row base
    {
        const int r = t >> 6, f = t & 63;
        hl[r][f] = h[(base + r) * F_ + f];
    }
    __syncthreads();

    const int r = t >> 6, f = t & 63;
    float acc = 0.0f;
#pragma unroll
    for (int k = 0; k < F_; ++k) acc = fmaf(hl[r][k], Wl[k * F_ + f], acc);

    ws[(base + r) * F_ + f] = acc;            // Wh
    red1[r][f] = acc * al[f];
    red2[r][f] = acc * al[F_ + f];
    __syncthreads();

    if (t < 8) {
        const int rr = t >> 1, which = t & 1;
        float s = 0.0f;
        for (int k = 0; k < F_; ++k) s += which ? red2[rr][k] : red1[rr][k];
        if (which) ws[S2_OFF + base + rr] = s;
        else       ws[S1_OFF + base + rr] = s;
    }
}

// ---------------------------------------------------------------------------
// Kernel 2: s2max[b] = max_j s2[b,j]   (lrelu is monotone -> row max is cheap)
// ---------------------------------------------------------------------------
__global__ void __launch_bounds__(256)
gat_s2max_kernel(float* __restrict__ ws) {
    __shared__ float mbuf[256];
    const int b = blockIdx.x, t = threadIdx.x;
    const float* s2 = ws + S2_OFF + (long)b * N_;
    float mx = -3.0e38f;
    for (int j = t; j < N_; j += 256) mx = fmaxf(mx, s2[j]);
    mbuf[t] = mx;
    __syncthreads();
    if (t == 0) {
        float r = mbuf[0];
        for (int i = 1; i < 256; ++i) r = fmaxf(r, mbuf[i]);
        ws[S2MAX_OFF + b] = r;
    }
}

// ---------------------------------------------------------------------------
// Kernel 3: fused softmax-attention GEMM.
//   out[b, i0:i0+16, :] = (P @ Wh) / (P @ ones),  P_ij = exp(lrelu(s1_i+s2_j)-m_i)
// One workgroup (8 wave32) per 16-row block; each wave owns 512 j's.
// A fragments (16x4 f32) are synthesized in registers from s2 staged in LDS;
// B fragments (4x16 f32) stream Wh columns from L2. 5 accumulator tiles:
// 4 for f=0..63 plus a ones-column tile that yields the softmax denominator.
// ---------------------------------------------------------------------------
__global__ void __launch_bounds__(256)
gat_attn_kernel(const float* __restrict__ ws, float* __restrict__ out) {
    __shared__ float s2l[N_];            // 16 KB
    __shared__ float part[8 * 5 * 256]; // 40 KB: per-wave partial C tiles
    __shared__ float fin[5 * 256];      // 5 KB: reduced tiles

    const int t    = threadIdx.x;
    const int lane = t & 31;
    const int wave = t >> 5;
    const int b    = blockIdx.y;
    const int i0   = blockIdx.x * 16;

    const float* Wh = ws + (long)b * N_ * F_;
    const float* s1 = ws + S1_OFF + (long)b * N_;
    const float* s2 = ws + S2_OFF + (long)b * N_;
    const float  s2max = ws[S2MAX_OFF + b];

    // Stage s2[b,:] into LDS
    for (int j = t; j < N_; j += 256) s2l[j] = s2[j];
    __syncthreads();

    // Per-lane constants (ISA 7.12.2 32-bit A layout: lane%16 = M, lane/16 picks K pair)
    const int   mrow_i = lane & 15;
    const int   khalf  = lane >> 4;
    const int   ncol   = lane & 15;
    const float s1i    = s1[i0 + mrow_i];
    const float zmax   = s1i + s2max;
    const float rowmax = fmaxf(zmax, 0.01f * zmax);   // lrelu(s1_i + max s2)
    const float LOG2E  = 1.4426950408889634f;

    v8f c0 = {}, c1 = {}, c2 = {}, c3 = {}, c4 = {};
    v2f bones;                                        // ones-column B tile
    bones.x = (ncol == 0) ? 1.0f : 0.0f;
    bones.y = bones.x;

    const int jstart = wave * (N_ / 8);               // 512 j's per wave
    for (int step = 0; step < (N_ / 8) / 4; ++step) { // K=4 per WMMA
        const int jk = jstart + step * 4 + khalf * 2;

        // --- A fragment: two attention weights per lane ---
        float e0 = s1i + s2l[jk];
        float e1 = s1i + s2l[jk + 1];
        e0 = fmaxf(e0, 0.01f * e0);                   // leaky_relu
        e1 = fmaxf(e1, 0.01f * e1);
        v2f A;
        A.x = EXP2F((e0 - rowmax) * LOG2E);
        A.y = EXP2F((e1 - rowmax) * LOG2E);

        // --- B fragments: Wh rows jk, jk+1 at cols tile*16 + ncol ---
        const float* r0 = Wh + (long)jk * F_;
        const float* r1 = r0 + F_;
        v2f B0, B1, B2, B3;
        B0.x = r0[ncol];      B0.y = r1[ncol];
        B1.x = r0[16 + ncol]; B1.y = r1[16 + ncol];
        B2.x = r0[32 + ncol]; B2.y = r1[32 + ncol];
        B3.x = r0[48 + ncol]; B3.y = r1[48 + ncol];

        c0 = __builtin_amdgcn_wmma_f32_16x16x4_f32(false, A, false, B0, (short)0, c0, false, false);
        c1 = __builtin_amdgcn_wmma_f32_16x16x4_f32(false, A, false, B1, (short)0, c1, false, false);
        c2 = __builtin_amdgcn_wmma_f32_16x16x4_f32(false, A, false, B2, (short)0, c2, false, false);
        c3 = __builtin_amdgcn_wmma_f32_16x16x4_f32(false, A, false, B3, (short)0, c3, false, false);
        c4 = __builtin_amdgcn_wmma_f32_16x16x4_f32(false, A, false, bones, (short)0, c4, false, false);
    }

    // Dump per-wave partial tiles to LDS
    float* pw = part + wave * (5 * 256);
#pragma unroll
    for (int v = 0; v < 8; ++v) {
        pw[0 * 256 + v * 32 + lane] = c0[v];
        pw[1 * 256 + v * 32 + lane] = c1[v];
        pw[2 * 256 + v * 32 + lane] = c2[v];
        pw[3 * 256 + v * 32 + lane] = c3[v];
        pw[4 * 256 + v * 32 + lane] = c4[v];
    }
    __syncthreads();

    // Cross-wave K reduction: 1280 tile elements, 8 partials each
    for (int r = 0; r < 5; ++r) {
        const int idx = r * 256 + t;
        float s = 0.0f;
#pragma unroll
        for (int w = 0; w < 8; ++w) s += part[w * 1280 + idx];
        fin[idx] = s;
    }
    __syncthreads();

    // Finalize: divide by fused denominator (tile 4, column 0) and store.
    // C/D layout: element (m,n) -> vgpr m%8, lane (m/8)*16 + n
#pragma unroll
    for (int it = 0; it < 4; ++it) {
        const int o  = it * 256 + t;      // 0..1023 = 16 rows x 64 cols
        const int mm = o >> 6;
        const int f  = o & 63;
        const int tile = f >> 4, nn = f & 15;
        const int cidx = (mm & 7) * 32 + (mm >> 3) * 16;
        const float num = fin[tile * 256 + cidx + nn];
        const float den = fin[4 * 256 + cidx];
        out[((long)b * N_ + i0 + mm) * F_ + f] = num / den;
    }
}

// ---------------------------------------------------------------------------
extern "C" void kernel_launch(void* const* d_in, const int* in_sizes, int n_in,
                              void* d_out, int out_size, void* d_ws, size_t ws_size,
                              hipStream_t stream) {
    const float* h   = (const float*)d_in[0];
    const float* adj = (const float*)d_in[1]; (void)adj;  // unused by forward
    const float* W   = (const float*)d_in[2];
    const float* a   = (const float*)d_in[3];
    float* out = (float*)d_out;
    float* ws  = (float*)d_ws;
    (void)in_sizes; (void)n_in; (void)out_size; (void)ws_size;

    gat_wh_kernel  <<<dim3(B_ * N_ / 4), 256, 0, stream>>>(h, W, a, ws);
    gat_s2max_kernel<<<dim3(B_),         256, 0, stream>>>(ws);
    gat_attn_kernel<<<dim3(N_ / 16, B_), 256, 0, stream>>>(ws, out);
}